// VITTransformer_89996744721129
// MI455X (gfx1250) — compile-verified
//
#include <hip/hip_runtime.h>
#include <hip/hip_bf16.h>
#include <math.h>

// ---------------------------------------------------------------------------
// Problem constants
// ---------------------------------------------------------------------------
#define BB 32
#define CC 256
#define SS 1024
#define MM (BB * SS)          // 32768 rows
#define LL 8
#define SLICE ((size_t)MM * CC)   // 8,388,608 floats = one (B,S,C) tensor

typedef __attribute__((ext_vector_type(2))) float v2f;
typedef __attribute__((ext_vector_type(8))) float v8f;

// ---------------------------------------------------------------------------
// Stem: u[c] = conv_w . feat_w[:,c] ; v[c] = conv_b . feat_w[:,c] + feat_b[c]
// ---------------------------------------------------------------------------
__global__ __launch_bounds__(256) void stem_uv_kernel(
    const float* __restrict__ conv_w, const float* __restrict__ conv_b,
    const float* __restrict__ feat_w, const float* __restrict__ feat_b,
    float* __restrict__ uv)
{
    int c = threadIdx.x;
    float u = 0.f, v = 0.f;
    for (int k = 0; k < CC; ++k) {
        float w = feat_w[k * CC + c];
        u += conv_w[k] * w;
        v += conv_b[k] * w;
    }
    uv[c]      = u;
    uv[CC + c] = v + feat_b[c];
}

// x[b,s,c] = tanh(image[b,0,16*(s/32),16*(s%32)] * u[c] + v[c])
__global__ __launch_bounds__(256) void stem_kernel(
    const float* __restrict__ image, const float* __restrict__ uv,
    float* __restrict__ X)
{
    int m = blockIdx.x;             // b*1024 + s
    int c = threadIdx.x;
    int b = m >> 10;
    int s = m & 1023;
    float xs = image[(size_t)b * 262144 + (size_t)(s >> 5) * 8192 + (size_t)(s & 31) * 16];
    X[(size_t)m * CC + c] = tanhf(xs * uv[c] + uv[CC + c]);
}

// ---------------------------------------------------------------------------
// LayerNorm over C=256: one wave per row, 8 elems per lane (wave32)
// ---------------------------------------------------------------------------
__global__ __launch_bounds__(256) void ln_kernel(
    const float* __restrict__ X, const float* __restrict__ ls,
    const float* __restrict__ lb, float* __restrict__ TN)
{
    int wave = threadIdx.x >> 5;
    int lane = threadIdx.x & 31;
    int m = blockIdx.x * 8 + wave;
    const float* row = X + (size_t)m * CC;
    float v[8];
    float sum = 0.f, sq = 0.f;
    #pragma unroll
    for (int j = 0; j < 8; ++j) {
        v[j] = row[j * 32 + lane];
        sum += v[j];
        sq  += v[j] * v[j];
    }
    #pragma unroll
    for (int o = 16; o >= 1; o >>= 1) {
        sum += __shfl_xor(sum, o, 32);
        sq  += __shfl_xor(sq,  o, 32);
    }
    float mu  = sum * (1.0f / CC);
    float var = sq * (1.0f / CC) - mu * mu;
    float rs  = rsqrtf(var + 1e-5f);
    float* out = TN + (size_t)m * CC;
    #pragma unroll
    for (int j = 0; j < 8; ++j) {
        int c = j * 32 + lane;
        out[c] = (v[j] - mu) * rs * ls[c] + lb[c];
    }
}

// ---------------------------------------------------------------------------
// GEMM: Y = act(X @ W + bias) [+ res].  X: M x 256, W: 256 x N (row major).
// Block: 256 threads (8 waves) -> tile 128(M) x 128(N), K chunk 32.
// Waves arranged 4(M) x 2(N); each wave computes 32(M) x 64(N) = 8
// accumulators of V_WMMA_F32_16X16X4_F32.  X and W chunks are staged into
// LDS with global_load_async_to_lds_b128 (ASYNCcnt path), so the inner
// loop feeds WMMA purely from LDS.
// ACT: 1 = sigmoid, 2 = relu.
// ---------------------------------------------------------------------------
#define BM 128
#define BN 128
#define KC 32
#define SXS 36    // sX row stride (dwords); 36 = 4*9, 9 coprime 16 -> no conflicts
#define SWS 132   // sW row stride (dwords); 132 % 64 = 4

template <int ACT, bool RES>
__global__ __launch_bounds__(256) void gemm_k256(
    const float* __restrict__ X, const float* __restrict__ W,
    const float* __restrict__ bias, const float* __restrict__ res,
    float* __restrict__ Y, int N)
{
    __shared__ float sX[BM * SXS];   // 18,432 B
    __shared__ float sW[KC * SWS];   // 16,896 B

    const int tid  = threadIdx.x;
    const int wave = tid >> 5;
    const int lane = tid & 31;
    const int m0 = blockIdx.x * BM;
    const int n0 = blockIdx.y * BN;

    const int mw    = (wave & 3) * 32;     // wave M offset in tile
    const int nw    = (wave >> 2) * 64;    // wave N offset in tile
    const int lrow  = lane & 15;           // M (for A) / N (for B,C,D) within 16
    const int khalf = (lane >> 4) * 2;     // +0 or +2 in K

    // LDS byte offsets of the staging buffers (flat shared ptr low 32 bits)
    const unsigned sx_base = (unsigned)(size_t)(&sX[0]);
    const unsigned sw_base = (unsigned)(size_t)(&sW[0]);

    v8f acc0[4] = {{}, {}, {}, {}};
    v8f acc1[4] = {{}, {}, {}, {}};

    for (int kk = 0; kk < 256; kk += KC) {
        // ---- async stage X[m0:m0+128, kk:kk+32] : 4 x b128 per thread ----
        #pragma unroll
        for (int i = 0; i < 4; ++i) {
            int idx = tid + i * 256;
            int r  = idx >> 3;            // 0..127
            int c4 = (idx & 7) << 2;      // 0,4,..,28
            unsigned goff = (unsigned)((((m0 + r) * 256) + kk + c4) * 4);
            unsigned loff = sx_base + (unsigned)((r * SXS + c4) * 4);
            asm volatile("global_load_async_to_lds_b128 %0, %1, %2"
                         :: "v"(loff), "v"(goff), "s"(X) : "memory");
        }
        // ---- async stage W[kk:kk+32, n0:n0+128] : 4 x b128 per thread ----
        #pragma unroll
        for (int i = 0; i < 4; ++i) {
            int idx = tid + i * 256;
            int r  = idx >> 5;            // 0..31
            int c4 = (idx & 31) << 2;     // 0,4,..,124
            unsigned goff = (unsigned)((((kk + r) * N) + n0 + c4) * 4);
            unsigned loff = sw_base + (unsigned)((r * SWS + c4) * 4);
            asm volatile("global_load_async_to_lds_b128 %0, %1, %2"
                         :: "v"(loff), "v"(goff), "s"(W) : "memory");
        }
        asm volatile("s_wait_asynccnt 0x0" ::: "memory");
        __syncthreads();

        // ---- compute: 8 k-steps x 8 WMMA, all operands from LDS ----
        const float* ax0 = &sX[(mw + lrow) * SXS + khalf];
        const float* ax1 = ax0 + 16 * SXS;
        #pragma unroll
        for (int k2 = 0; k2 < KC; k2 += 4) {
            v2f a0, a1;
            a0.x = ax0[k2];
            a0.y = ax0[k2 + 1];
            a1.x = ax1[k2];
            a1.y = ax1[k2 + 1];
            const float* w0 = &sW[(k2 + khalf) * SWS + nw + lrow];
            const float* w1 = w0 + SWS;
            #pragma unroll
            for (int t = 0; t < 4; ++t) {
                v2f b;
                b.x = w0[t * 16];
                b.y = w1[t * 16];
                acc0[t] = __builtin_amdgcn_wmma_f32_16x16x4_f32(false, a0, false, b, (short)0, acc0[t], false, false);
                acc1[t] = __builtin_amdgcn_wmma_f32_16x16x4_f32(false, a1, false, b, (short)0, acc1[t], false, false);
            }
        }
        __syncthreads();
    }

    // Epilogue.  C/D layout: VGPR r, lanes 0-15 -> (M=r,   N=lane)
    //                                lanes 16-31 -> (M=r+8, N=lane-16)
    const int mtop = (lane >> 4) << 3;
    #pragma unroll
    for (int mi = 0; mi < 2; ++mi) {
        const v8f* accs = mi ? acc1 : acc0;
        const int mrow0 = m0 + mw + mi * 16 + mtop;
        #pragma unroll
        for (int t = 0; t < 4; ++t) {
            int col = n0 + nw + t * 16 + lrow;
            float bb = bias[col];
            #pragma unroll
            for (int r = 0; r < 8; ++r) {
                float val = accs[t][r] + bb;
                if (ACT == 1) val = 1.0f / (1.0f + expf(-val));
                if (ACT == 2) val = fmaxf(val, 0.0f);
                size_t idx = (size_t)(mrow0 + r) * N + col;
                if (RES) val += res[idx];
                Y[idx] = val;
            }
        }
    }
}

// ---------------------------------------------------------------------------
// Attention: one block (256 threads) per (b,c) channel; 32x32 spatial tile.
// Fuses residual: TA = X + attn_out.
// ---------------------------------------------------------------------------
__global__ __launch_bounds__(256) void attn_kernel(
    const float* __restrict__ Qg, const float* __restrict__ Kg,
    const float* __restrict__ Vg, const float* __restrict__ Xg,
    float* __restrict__ TA)
{
    __shared__ float sQ[32 * 33], sK[32 * 33], sV[32 * 33];
    __shared__ float skp[8 * 32], sqp[8 * 32];
    __shared__ float sQk[32 * 8];   // softmax(Q @ kp^T)
    __shared__ float sKq[8 * 32];   // softmax(qp @ K^T)
    __shared__ float sKqv[8 * 33];

    const int tid = threadIdx.x;
    const int b = blockIdx.x >> 8;      // /C
    const int c = blockIdx.x & 255;     // %C

    for (int idx = tid; idx < 1024; idx += 256) {
        size_t g = ((size_t)(b * 1024 + idx)) * CC + c;
        int li = (idx >> 5) * 33 + (idx & 31);
        sQ[li] = Qg[g];
        sK[li] = Kg[g];
        sV[li] = Vg[g];
    }
    __syncthreads();

    // pool over rows (i) in groups of 4:  kp/qp : [8][32]
    {
        int jp = tid >> 5, k = tid & 31;
        int r0 = (4 * jp) * 33 + k;
        skp[jp * 32 + k] = 0.25f * (sK[r0] + sK[r0 + 33] + sK[r0 + 66] + sK[r0 + 99]);
        sqp[jp * 32 + k] = 0.25f * (sQ[r0] + sQ[r0 + 33] + sQ[r0 + 66] + sQ[r0 + 99]);
    }
    __syncthreads();

    const float d = 0.0625f;   // 1/sqrt(256)

    // Qk[i, jp] = softmax_jp( d * sum_k Q[i,k] * kp[jp,k] ),  i:32, jp:8
    {
        int i = tid >> 3, jp = tid & 7;
        float acc = 0.f;
        #pragma unroll 8
        for (int k = 0; k < 32; ++k) acc += sQ[i * 33 + k] * skp[jp * 32 + k];
        acc *= d;
        float mx = acc;
        #pragma unroll
        for (int o = 4; o >= 1; o >>= 1) mx = fmaxf(mx, __shfl_xor(mx, o, 8));
        float e = expf(acc - mx);
        float sm = e;
        #pragma unroll
        for (int o = 4; o >= 1; o >>= 1) sm += __shfl_xor(sm, o, 8);
        sQk[i * 8 + jp] = e / sm;
    }
    // Kq[jp, j] = softmax_j( d * sum_k qp[jp,k] * K[j,k] ),  jp:8, j:32
    {
        int jp = tid >> 5, j = tid & 31;
        float acc = 0.f;
        #pragma unroll 8
        for (int k = 0; k < 32; ++k) acc += sqp[jp * 32 + k] * sK[j * 33 + k];
        acc *= d;
        float mx = acc;
        #pragma unroll
        for (int o = 16; o >= 1; o >>= 1) mx = fmaxf(mx, __shfl_xor(mx, o, 32));
        float e = expf(acc - mx);
        float sm = e;
        #pragma unroll
        for (int o = 16; o >= 1; o >>= 1) sm += __shfl_xor(sm, o, 32);
        sKq[jp * 32 + j] = e / sm;
    }
    __syncthreads();

    // Kqv[jp, j] = sum_k Kq[jp,k] * V[k,j]
    {
        int jp = tid >> 5, j = tid & 31;
        float acc = 0.f;
        #pragma unroll 8
        for (int k = 0; k < 32; ++k) acc += sKq[jp * 32 + k] * sV[k * 33 + j];
        sKqv[jp * 33 + j] = acc;
    }
    __syncthreads();

    // out[i, j] = sum_k Qk[i,k] * Kqv[k,j];  TA = X + out  (s = i*32 + j)
    {
        int j = tid & 31;
        int ib = (tid >> 5) * 4;
        #pragma unroll
        for (int r = 0; r < 4; ++r) {
            int i = ib + r;
            float o = 0.f;
            #pragma unroll
            for (int k = 0; k < 8; ++k) o += sQk[i * 8 + k] * sKqv[k * 33 + j];
            size_t g = ((size_t)(b * 1024 + i * 32 + j)) * CC + c;
            TA[g] = Xg[g] + o;
        }
    }
}

// ---------------------------------------------------------------------------
// Gate: tn2[m,c] = g[m,2c] * ta[m,c] + g[m,2c+1]
// ---------------------------------------------------------------------------
__global__ __launch_bounds__(256) void gate_kernel(
    const float* __restrict__ G, const float* __restrict__ TA,
    float* __restrict__ TN2)
{
    size_t m = blockIdx.x;
    int c = threadIdx.x;
    float2 g2 = ((const float2*)(G + m * 512))[c];
    TN2[m * CC + c] = g2.x * TA[m * CC + c] + g2.y;
}

// ---------------------------------------------------------------------------
// Host launch
// ---------------------------------------------------------------------------
extern "C" void kernel_launch(void* const* d_in, const int* in_sizes, int n_in,
                              void* d_out, int out_size, void* d_ws, size_t ws_size,
                              hipStream_t stream)
{
    const float* image  = (const float*)d_in[0];
    const float* conv_w = (const float*)d_in[1];
    const float* conv_b = (const float*)d_in[2];
    const float* feat_w = (const float*)d_in[3];
    const float* feat_b = (const float*)d_in[4];
    const float* ln_s   = (const float*)d_in[5];
    const float* ln_b   = (const float*)d_in[6];
    const float* wq     = (const float*)d_in[7];
    const float* bq     = (const float*)d_in[8];
    const float* wk     = (const float*)d_in[9];
    const float* bk     = (const float*)d_in[10];
    const float* wv     = (const float*)d_in[11];
    const float* bv     = (const float*)d_in[12];
    const float* wf     = (const float*)d_in[13];
    const float* bf     = (const float*)d_in[14];
    const float* wa     = (const float*)d_in[15];
    const float* ba     = (const float*)d_in[16];

    float* ws  = (float*)d_ws;
    float* X   = ws + 0 * SLICE;
    float* TN  = ws + 1 * SLICE;
    float* Qb  = ws + 2 * SLICE;
    float* Kb  = ws + 3 * SLICE;
    float* Vb  = ws + 4 * SLICE;
    float* TA  = ws + 5 * SLICE;
    float* UV  = ws + 6 * SLICE;
    float* G   = Kb;   // 2*SLICE scratch (K,V region free when used)
    float* TN2 = Qb;   // Q region free when used
    float* out = (float*)d_out;

    // stem
    stem_uv_kernel<<<1, 256, 0, stream>>>(conv_w, conv_b, feat_w, feat_b, UV);
    stem_kernel<<<MM, 256, 0, stream>>>(image, UV, X);

    dim3 g256(MM / BM, 2);   // N = 256
    dim3 g512(MM / BM, 4);   // N = 512

    for (int l = 0; l < LL; ++l) {
        ln_kernel<<<MM / 8, 256, 0, stream>>>(X, ln_s + l * CC, ln_b + l * CC, TN);

        gemm_k256<1, false><<<g256, 256, 0, stream>>>(TN, wq + (size_t)l * CC * CC, bq + l * CC, nullptr, Qb, 256);
        gemm_k256<1, false><<<g256, 256, 0, stream>>>(TN, wk + (size_t)l * CC * CC, bk + l * CC, nullptr, Kb, 256);
        gemm_k256<1, false><<<g256, 256, 0, stream>>>(TN, wv + (size_t)l * CC * CC, bv + l * CC, nullptr, Vb, 256);

        attn_kernel<<<BB * CC, 256, 0, stream>>>(Qb, Kb, Vb, X, TA);

        gemm_k256<1, false><<<g512, 256, 0, stream>>>(TA, wa + (size_t)l * CC * 512, ba + l * 512, nullptr, G, 512);
        gate_kernel<<<MM, 256, 0, stream>>>(G, TA, TN2);

        gemm_k256<2, true><<<g256, 256, 0, stream>>>(TN2, wf + (size_t)l * CC * CC, bf + l * CC, TA, X, 256);

        if (l == 2 || l == 4 || l == 6) {
            int slot = l / 2;             // 1, 2, 3
            hipMemcpyAsync(out + (size_t)slot * SLICE, X, SLICE * sizeof(float),
                           hipMemcpyDeviceToDevice, stream);
            hipMemcpyAsync(out + (size_t)(slot + 3) * SLICE, TA, SLICE * sizeof(float),
                           hipMemcpyDeviceToDevice, stream);
        }
    }
    hipMemcpyAsync(out, X, SLICE * sizeof(float), hipMemcpyDeviceToDevice, stream);
}